// MemAttention_8710193676604
// MI455X (gfx1250) — compile-verified
//
#include <hip/hip_runtime.h>
#include <math.h>

// CDNA5 / gfx1250, wave32. WMMA f32 16x16x4: per-lane A,B = 2 VGPRs (v2f), C/D = 8 VGPRs (v8f).
typedef __attribute__((ext_vector_type(2))) float v2f;
typedef __attribute__((ext_vector_type(8))) float v8f;

#define NPIX   1024           // H*W
#define JC     16             // j values staged per chunk (8 even/odd pairs)
// Pair-interleaved LDS layouts: the two K components of a WMMA A/B operand sit
// 33 dwords apart from one base -> single ds_load_2addr_b32 offset1:33.
// sW: [jpair(8)][m(16)][66]  element (jl,ml,pp) at (jl>>1 *16 + ml)*66 + (jl&1)*33 + pp
// sX: [b(4)][jpair(8)][66]   element (b,jl,pp)  at (b*8 + (jl>>1))*66 + (jl&1)*33 + pp
#define SW_FLOATS (8 * 16 * 66)
#define SX_FLOATS (4 * 8 * 66)

__device__ __forceinline__ void async_ld_b32(unsigned lds_off, const float* gaddr) {
    // GLOBAL_LOAD_ASYNC_TO_LDS_B32: per-lane global -> LDS, tracked by ASYNCcnt.
    asm volatile("global_load_async_to_lds_b32 %0, %1, off"
                 :: "v"(lds_off), "v"(gaddr) : "memory");
}

// ---------------------------------------------------------------------------
// Projection: Y[b, m, px] = sum_j X[b, j, px] * W[j, m, px]    (per-pixel GEMM)
// grid.x = pixel-block (32 px), grid.y = m-tile (16 m). block = 256 (8 waves).
// Wave w owns pixels {4w..4w+3}; D-tile columns 0..3 = batch b.
// ---------------------------------------------------------------------------
__global__ __launch_bounds__(256)
void proj_wmma_kernel(const float* __restrict__ X, const float* __restrict__ W,
                      float* __restrict__ Y, int Dj, int Dm)
{
    __shared__ float sW[SW_FLOATS];
    __shared__ float sX[SX_FLOATS];

    const int tid  = threadIdx.x;
    const int lane = tid & 31;
    const int wv   = tid >> 5;        // 0..7
    const int mL   = lane & 15;       // m-row within tile / batch column
    const int hf   = lane >> 4;       // half-wave select
    const int px0  = blockIdx.x * 32;
    const int m0   = blockIdx.y * 16;

    v8f acc[4];
    #pragma unroll
    for (int i = 0; i < 4; ++i)
        #pragma unroll
        for (int e = 0; e < 8; ++e) acc[i][e] = 0.0f;

    const int nchunks = Dj / JC;
    for (int jc = 0; jc < nchunks; ++jc) {
        const int j0 = jc * JC;

        // ---- async-stage W[j0..j0+15, m0..m0+15, px0..px0+31] (coalesced) ----
        #pragma unroll
        for (int it = 0; it < 32; ++it) {
            int e  = it * 256 + tid;          // 0..8191
            int pp = e & 31;
            int ml = (e >> 5) & 15;
            int jl = e >> 9;                  // 0..15
            const float* gp = W + ((size_t)(j0 + jl) * Dm + (m0 + ml)) * NPIX + px0 + pp;
            unsigned lo = (unsigned)(size_t)(sW + ((jl >> 1) * 16 + ml) * 66 + (jl & 1) * 33 + pp);
            async_ld_b32(lo, gp);
        }
        // ---- async-stage X[0..3, j0..j0+15, px0..px0+31] ----
        #pragma unroll
        for (int it = 0; it < 8; ++it) {
            int e  = it * 256 + tid;          // 0..2047
            int pp = e & 31;
            int jl = (e >> 5) & 15;
            int b  = e >> 9;                  // 0..3
            const float* gp = X + ((size_t)b * Dj + (j0 + jl)) * NPIX + px0 + pp;
            unsigned lo = (unsigned)(size_t)(sX + (b * 8 + (jl >> 1)) * 66 + (jl & 1) * 33 + pp);
            async_ld_b32(lo, gp);
        }
        asm volatile("s_wait_asynccnt 0" ::: "memory");
        __syncthreads();

        const int bb = (mL < 4) ? mL : 0;     // D columns 4..15 are don't-care
        #pragma unroll
        for (int i = 0; i < 4; ++i) {
            const int pp = wv * 4 + i;        // pixel within block
            #pragma unroll
            for (int s = 0; s < 4; ++s) {     // 4-wide j sub-chunks
                const int kp = s * 2 + hf;    // j-pair index; K = r + 2*hf handled by pairing
                const int abase = (kp * 16 + mL) * 66 + pp;
                const int bbase = (bb * 8 + kp) * 66 + pp;
                v2f av, bv;
                av.x = sW[abase];       av.y = sW[abase + 33];   // A[m, k], A[m, k+1]
                bv.x = sX[bbase];       bv.y = sX[bbase + 33];   // B[k, n=b], B[k+1, n=b]
                acc[i] = __builtin_amdgcn_wmma_f32_16x16x4_f32(
                    false, av, false, bv, (short)0, acc[i], false, false);
            }
        }
        __syncthreads();
    }

    // D layout: VGPR r -> row M = r + 8*hf, column N = mL (N<4 valid = batch b)
    if (mL < 4) {
        #pragma unroll
        for (int i = 0; i < 4; ++i) {
            const int px = px0 + wv * 4 + i;
            #pragma unroll
            for (int r = 0; r < 8; ++r) {
                const int m = m0 + r + 8 * hf;
                Y[((size_t)mL * Dm + m) * NPIX + px] = acc[i][r];
            }
        }
    }
}

// ---------------------------------------------------------------------------
// Attention: scores (m-reduction split across 8 waves, LDS-reduced), softmax,
// weighted read of values, residual add. Block = 256 threads / 32 pixels.
// ---------------------------------------------------------------------------
__global__ __launch_bounds__(256)
void attn_kernel(const float* __restrict__ fm, const float* __restrict__ qw,
                 const float* __restrict__ kw, const float* __restrict__ vw,
                 const float* __restrict__ kbuf, const float* __restrict__ vbuf,
                 float* __restrict__ out)
{
    __shared__ float sPart[8 * 48 * 32];   // [wave][b*12+n][px]
    __shared__ float sFin[48 * 32];        // [b*12+n][px]

    const int tid  = threadIdx.x;
    const int lane = tid & 31;
    const int wv   = tid >> 5;
    const int px   = blockIdx.x * 32 + lane;

    float s[4][12];
    #pragma unroll
    for (int b = 0; b < 4; ++b)
        #pragma unroll
        for (int n = 0; n < 12; ++n) s[b][n] = 0.0f;

    const int mBeg = wv * 32;
    #pragma unroll 2
    for (int m = mBeg; m < mBeg + 32; ++m) {
        float q[4];
        #pragma unroll
        for (int b = 0; b < 4; ++b) q[b] = qw[((size_t)b * 256 + m) * NPIX + px];
        #pragma unroll
        for (int n = 0; n < 4; ++n) {
            float kv = kw[((size_t)n * 256 + m) * NPIX + px];
            #pragma unroll
            for (int b = 0; b < 4; ++b) s[b][n] += q[b] * kv;
        }
        #pragma unroll
        for (int n = 0; n < 8; ++n) {
            float kv = kbuf[((size_t)n * 256 + m) * NPIX + px];
            #pragma unroll
            for (int b = 0; b < 4; ++b) s[b][4 + n] += q[b] * kv;
        }
    }
    #pragma unroll
    for (int b = 0; b < 4; ++b)
        #pragma unroll
        for (int n = 0; n < 12; ++n)
            sPart[(wv * 48 + b * 12 + n) * 32 + lane] = s[b][n];
    __syncthreads();

    #pragma unroll
    for (int pi = 0; pi < 6; ++pi) {
        const int pair = wv * 6 + pi;
        float a = 0.0f;
        #pragma unroll
        for (int w = 0; w < 8; ++w) a += sPart[(w * 48 + pair) * 32 + lane];
        sFin[pair * 32 + lane] = a;
    }
    __syncthreads();

    float attn[4][12];
    #pragma unroll
    for (int b = 0; b < 4; ++b) {
        float mx = -1e30f;
        #pragma unroll
        for (int n = 0; n < 12; ++n) {
            attn[b][n] = sFin[(b * 12 + n) * 32 + lane] * 0.0625f;   // 1/sqrt(256)
            mx = fmaxf(mx, attn[b][n]);
        }
        float sum = 0.0f;
        #pragma unroll
        for (int n = 0; n < 12; ++n) { attn[b][n] = __expf(attn[b][n] - mx); sum += attn[b][n]; }
        const float inv = 1.0f / sum;
        #pragma unroll
        for (int n = 0; n < 12; ++n) attn[b][n] *= inv;
    }

    #pragma unroll
    for (int dd = 0; dd < 8; ++dd) {
        const int d = wv * 8 + dd;
        float vv[12];
        #pragma unroll
        for (int n = 0; n < 4; ++n) vv[n]     = vw[((size_t)n * 64 + d) * NPIX + px];
        #pragma unroll
        for (int n = 0; n < 8; ++n) vv[4 + n] = vbuf[((size_t)n * 64 + d) * NPIX + px];
        #pragma unroll
        for (int b = 0; b < 4; ++b) {
            float r = 0.0f;
            #pragma unroll
            for (int n = 0; n < 12; ++n) r += attn[b][n] * vv[n];
            const size_t o = ((size_t)b * 64 + d) * NPIX + px;
            out[o] = fm[o] + 0.5f * r;
        }
    }
}

// ---------------------------------------------------------------------------
extern "C" void kernel_launch(void* const* d_in, const int* in_sizes, int n_in,
                              void* d_out, int out_size, void* d_ws, size_t ws_size,
                              hipStream_t stream) {
    const float* fc = (const float*)d_in[0];   // [4,256,32,32]
    const float* fm = (const float*)d_in[1];   // [4,64,32,32]
    const float* Q  = (const float*)d_in[2];   // [256,256,32,32]
    const float* K  = (const float*)d_in[3];   // [256,256,32,32]
    const float* V  = (const float*)d_in[4];   // [64,64,32,32]
    const float* kb = (const float*)d_in[5];   // [8,256,32,32]
    const float* vb = (const float*)d_in[6];   // [8,64,32,32]
    float* out = (float*)d_out;

    // workspace: q (4MB) + k (4MB) + v (1MB) = 9MB
    float* qw = (float*)d_ws;
    float* kw = qw + (size_t)4 * 256 * NPIX;
    float* vw = kw + (size_t)4 * 256 * NPIX;

    dim3 blk(256);
    proj_wmma_kernel<<<dim3(32, 16), blk, 0, stream>>>(fc, Q, qw, 256, 256);
    proj_wmma_kernel<<<dim3(32, 16), blk, 0, stream>>>(fc, K, kw, 256, 256);
    proj_wmma_kernel<<<dim3(32, 4),  blk, 0, stream>>>(fm, V, vw, 64, 64);
    attn_kernel<<<dim3(32), blk, 0, stream>>>(fm, qw, kw, vw, kb, vb, out);
}